// HM_29351806501221
// MI455X (gfx1250) — compile-verified
//
#include <hip/hip_runtime.h>
#include <math.h>

typedef float v2f __attribute__((ext_vector_type(2)));
typedef float v8f __attribute__((ext_vector_type(8)));

#define HSTRIDE 36          // 32 + 4 pad: conflict-free LDS transpose, 16B-aligned rows
#define WPB 8               // waves per block (256 threads, wave32)

// ---- d_ws layout (floats): [0..1] accumulators, [16..] uniform constant block ----
// (all offsets are multiples of 4 -> float4-aligned relative to ws+16, which is 64B aligned)
#define OFF_TL1   0      // 4x2  logits l1 = g4(z2)
#define OFF_SL1   8      // 4    softplus-sum of l1
#define OFF_TL3   12     // 4x2  logits l3 = g1(z4)
#define OFF_SL3   20     // 4
#define OFF_TP4   24     // 4x2  pz4 = sigmoid(inf3(z3))
#define OFF_TMU1  32     // 4    muy1 = g5(z1)
#define OFF_TMU2  36     // 4    muy2 = g2(z3)
#define OFF_WZ1A  40     // 8    inf1_w1 row 1 (z1a)
#define OFF_WZ1B  48     // 8    inf1_w1 row 2 (z1b)
#define OFF_W2I0  56     // 16   inf0_w2
#define OFF_W2I2  72     // 16   inf2_w2
#define OFF_W2G3  88     // 16   g3_w2
#define OFF_W2I1  104    // 16   inf1_w2
#define OFF_B2    120    // 8    {inf0_b2[2], inf2_b2[2], g3_b2[2], inf1_b2[2]}
#define OFF_L4    128    // 2    g0_w
#define OFF_WMAT  132    // 4x32 fused layer-1 weight matrix (rows: y1,y2,1,0)
#define CST_FLOATS 260   // total constant floats (ws needs (16+260)*4 = 1104 B)

struct HmParams {
  const float *y1,*y2,*n1,*n2,*n3,*n4;
  const float *inf0_w1,*inf0_b1,*inf0_w2,*inf0_b2;
  const float *inf1_w1,*inf1_b1,*inf1_w2,*inf1_b2;
  const float *inf2_w1,*inf2_b1,*inf2_w2,*inf2_b2;
  const float *inf3_w1,*inf3_b1,*inf3_w2,*inf3_b2;
  const float *g1_w1,*g1_b1,*g1_w2,*g1_b2;
  const float *g2_w1,*g2_b1,*g2_w2,*g2_b2;
  const float *g3_w1,*g3_b1,*g3_w2,*g3_b2;
  const float *g4_w1,*g4_b1,*g4_w2,*g4_b2;
  const float *g5_w1,*g5_b1,*g5_w2,*g5_b2;
  const float *g0_w;
  float *ws;
  int B;
};

__device__ __forceinline__ float sigf(float x){ return 1.0f/(1.0f+__expf(-x)); }
// max(l,0) + log1p(exp(-|l|))  (stable BCE softplus part)
__device__ __forceinline__ float spabs(float l){
  return fmaxf(l,0.0f) + __logf(1.0f+__expf(-fabsf(l)));
}
__device__ __forceinline__ float sel4f(bool a, bool b, float t0,float t1,float t2,float t3){
  float lo = a ? t1 : t0;
  float hi = a ? t3 : t2;
  return b ? hi : lo;
}

// ===================== prep kernel: uniform constants, computed ONCE =====================
__device__ void prep_mlp2(const float* w1,const float* b1,const float* w2,const float* b2,
                          float za,float zb,float* out){
  float o0=b2[0], o1=b2[1];
  #pragma clang loop unroll(disable)
  for (int j=0;j<8;++j){
    float h = fmaxf(b1[j] + za*w1[j] + zb*w1[8+j], 0.0f);
    o0 += h*w2[2*j]; o1 += h*w2[2*j+1];
  }
  out[0]=o0; out[1]=o1;
}
__device__ float prep_mlp1(const float* w1,const float* b1,const float* w2,const float* b2,
                           float za,float zb){
  float o0=b2[0];
  #pragma clang loop unroll(disable)
  for (int j=0;j<8;++j){
    float h = fmaxf(b1[j] + za*w1[j] + zb*w1[8+j], 0.0f);
    o0 += h*w2[j];
  }
  return o0;
}

__global__ void hm_prep(HmParams p){
  if (threadIdx.x != 0) return;
  p.ws[0]=0.0f; p.ws[1]=0.0f;
  float* C = p.ws + 16;
  #pragma clang loop unroll(disable)
  for (int q=0;q<4;++q){
    const float za=(float)(q&1), zb=(float)(q>>1);
    float o[2];
    prep_mlp2(p.g4_w1,p.g4_b1,p.g4_w2,p.g4_b2, za,zb,o);       // l1 = g4(z2)
    C[OFF_TL1+2*q]=o[0]; C[OFF_TL1+2*q+1]=o[1];
    C[OFF_SL1+q]=spabs(o[0])+spabs(o[1]);
    prep_mlp2(p.g1_w1,p.g1_b1,p.g1_w2,p.g1_b2, za,zb,o);       // l3 = g1(z4)
    C[OFF_TL3+2*q]=o[0]; C[OFF_TL3+2*q+1]=o[1];
    C[OFF_SL3+q]=spabs(o[0])+spabs(o[1]);
    prep_mlp2(p.inf3_w1,p.inf3_b1,p.inf3_w2,p.inf3_b2, za,zb,o); // pz4
    C[OFF_TP4+2*q]=sigf(o[0]); C[OFF_TP4+2*q+1]=sigf(o[1]);
    C[OFF_TMU1+q]=prep_mlp1(p.g5_w1,p.g5_b1,p.g5_w2,p.g5_b2, za,zb);
    C[OFF_TMU2+q]=prep_mlp1(p.g2_w1,p.g2_b1,p.g2_w2,p.g2_b2, za,zb);
  }
  #pragma clang loop unroll(disable)
  for (int j=0;j<8;++j){ C[OFF_WZ1A+j]=p.inf1_w1[8+j]; C[OFF_WZ1B+j]=p.inf1_w1[16+j]; }
  #pragma clang loop unroll(disable)
  for (int j=0;j<16;++j){
    C[OFF_W2I0+j]=p.inf0_w2[j]; C[OFF_W2I2+j]=p.inf2_w2[j];
    C[OFF_W2G3+j]=p.g3_w2[j];   C[OFF_W2I1+j]=p.inf1_w2[j];
  }
  C[OFF_B2+0]=p.inf0_b2[0]; C[OFF_B2+1]=p.inf0_b2[1];
  C[OFF_B2+2]=p.inf2_b2[0]; C[OFF_B2+3]=p.inf2_b2[1];
  C[OFF_B2+4]=p.g3_b2[0];   C[OFF_B2+5]=p.g3_b2[1];
  C[OFF_B2+6]=p.inf1_b2[0]; C[OFF_B2+7]=p.inf1_b2[1];
  C[OFF_L4+0]=p.g0_w[0];    C[OFF_L4+1]=p.g0_w[1];
  // fused layer-1 W[4x32]: rows = {y1, y2, bias(1), 0}; cols = [inf0|inf2|g3|inf1]
  #pragma clang loop unroll(disable)
  for (int col=0; col<32; ++col){
    float r0, r1, r2;
    if (col<8)       { r0=p.inf0_w1[col]; r1=p.inf0_w1[8+col]; r2=p.inf0_b1[col]; }
    else if (col<16) { r0=0.0f; r1=p.inf2_w1[col-8];  r2=p.inf2_b1[col-8]; }
    else if (col<24) { r0=0.0f; r1=p.g3_w1[col-16];   r2=p.g3_b1[col-16]; }
    else             { r0=0.0f; r1=p.inf1_w1[col-24]; r2=p.inf1_b1[col-24]; }
    C[OFF_WMAT+0*32+col]=r0; C[OFF_WMAT+1*32+col]=r1;
    C[OFF_WMAT+2*32+col]=r2; C[OFF_WMAT+3*32+col]=0.0f;
  }
}

// ===================== main kernel =====================
struct MainArgs {
  const float *y1,*y2,*n1,*n2,*n3,*n4;
  const float* __restrict__ cst;   // ws+16 (read-only in this kernel)
  float* __restrict__ acc;         // ws+0  (atomics only)
  int B;
};

__device__ __forceinline__ void dot8(float4 a, float4 b, const float* w2,
                                     float bias0, float bias1, float& o0, float& o1){
  float hh[8]={a.x,a.y,a.z,a.w,b.x,b.y,b.z,b.w};
  float s0=bias0, s1=bias1;
  #pragma unroll
  for (int j=0;j<8;++j){ float h=fmaxf(hh[j],0.0f); s0+=h*w2[2*j]; s1+=h*w2[2*j+1]; }
  o0=s0; o1=s1;
}
__device__ __forceinline__ void dot8z(float4 a, float4 b, const float* w2,
                                      float bias0, float bias1,
                                      const float* wza, const float* wzb,
                                      float z1a, float z1b, float& o0, float& o1){
  float hh[8]={a.x,a.y,a.z,a.w,b.x,b.y,b.z,b.w};
  float s0=bias0, s1=bias1;
  #pragma unroll
  for (int j=0;j<8;++j){
    float h=fmaxf(hh[j] + z1a*wza[j] + z1b*wzb[j], 0.0f);
    s0+=h*w2[2*j]; s1+=h*w2[2*j+1];
  }
  o0=s0; o1=s1;
}

__global__ __launch_bounds__(256) void hm_main(MainArgs p){
  const int lane   = threadIdx.x & 31;
  const int wv     = threadIdx.x >> 5;
  const int gw     = blockIdx.x * WPB + wv;
  const int nw     = gridDim.x * WPB;
  const int nTiles = p.B >> 5;                 // 32 samples per wave-iteration
  const bool hiH   = lane >= 16;
  const int  ncol  = lane & 15;

  __shared__ float Hs[WPB*32*HSTRIDE];
  __shared__ float redZ, redY;
  if (threadIdx.x==0){ redZ=0.0f; redY=0.0f; }
  __syncthreads();

  const float* __restrict__ C = p.cst;
  #define LD4(o) (*(const float4*)(C + (o)))
  // ---- hoist all uniform constants into registers (vectorized one-time loads) ----
  float tl10[4],tl11[4],sl1v[4],tl30[4],tl31[4],sl3v[4],tp40[4],tp41[4],tm1[4],tm2[4];
  {
    float4 u0=LD4(OFF_TL1), u1=LD4(OFF_TL1+4);
    tl10[0]=u0.x; tl11[0]=u0.y; tl10[1]=u0.z; tl11[1]=u0.w;
    tl10[2]=u1.x; tl11[2]=u1.y; tl10[3]=u1.z; tl11[3]=u1.w;
    float4 us=LD4(OFF_SL1); sl1v[0]=us.x; sl1v[1]=us.y; sl1v[2]=us.z; sl1v[3]=us.w;
    float4 v0=LD4(OFF_TL3), v1=LD4(OFF_TL3+4);
    tl30[0]=v0.x; tl31[0]=v0.y; tl30[1]=v0.z; tl31[1]=v0.w;
    tl30[2]=v1.x; tl31[2]=v1.y; tl30[3]=v1.z; tl31[3]=v1.w;
    float4 vs=LD4(OFF_SL3); sl3v[0]=vs.x; sl3v[1]=vs.y; sl3v[2]=vs.z; sl3v[3]=vs.w;
    float4 w0=LD4(OFF_TP4), w1=LD4(OFF_TP4+4);
    tp40[0]=w0.x; tp41[0]=w0.y; tp40[1]=w0.z; tp41[1]=w0.w;
    tp40[2]=w1.x; tp41[2]=w1.y; tp40[3]=w1.z; tp41[3]=w1.w;
    float4 m1=LD4(OFF_TMU1); tm1[0]=m1.x; tm1[1]=m1.y; tm1[2]=m1.z; tm1[3]=m1.w;
    float4 m2=LD4(OFF_TMU2); tm2[0]=m2.x; tm2[1]=m2.y; tm2[2]=m2.z; tm2[3]=m2.w;
  }
  float wza[8], wzb[8];
  #pragma unroll
  for (int k=0;k<2;++k){
    float4 a=LD4(OFF_WZ1A+4*k), b=LD4(OFF_WZ1B+4*k);
    wza[4*k]=a.x; wza[4*k+1]=a.y; wza[4*k+2]=a.z; wza[4*k+3]=a.w;
    wzb[4*k]=b.x; wzb[4*k+1]=b.y; wzb[4*k+2]=b.z; wzb[4*k+3]=b.w;
  }
  float w2i0[16], w2i2[16], w2g3[16], w2i1[16];
  #pragma unroll
  for (int k=0;k<4;++k){
    float4 a=LD4(OFF_W2I0+4*k), b=LD4(OFF_W2I2+4*k), c=LD4(OFF_W2G3+4*k), d=LD4(OFF_W2I1+4*k);
    w2i0[4*k]=a.x; w2i0[4*k+1]=a.y; w2i0[4*k+2]=a.z; w2i0[4*k+3]=a.w;
    w2i2[4*k]=b.x; w2i2[4*k+1]=b.y; w2i2[4*k+2]=b.z; w2i2[4*k+3]=b.w;
    w2g3[4*k]=c.x; w2g3[4*k+1]=c.y; w2g3[4*k+2]=c.z; w2g3[4*k+3]=c.w;
    w2i1[4*k]=d.x; w2i1[4*k+1]=d.y; w2i1[4*k+2]=d.z; w2i1[4*k+3]=d.w;
  }
  float4 bb0=LD4(OFF_B2), bb1=LD4(OFF_B2+4);
  const float b2i00=bb0.x, b2i01=bb0.y, b2i20=bb0.z, b2i21=bb0.w;
  const float b2g30=bb1.x, b2g31=bb1.y, b2i10=bb1.z, b2i11=bb1.w;
  float2 l4v = *(const float2*)(C + OFF_L4);
  const float l4_0=l4v.x, l4_1=l4v.y;

  // B operand of the fused layer-1 WMMA (per-lane gather):
  // VGPR0 = {row0 lo-lanes | row2 hi-lanes}, VGPR1 = {row1 | row3}
  const int kb = hiH ? 2 : 0;
  v2f bLo = { C[OFF_WMAT + kb*32 + ncol],      C[OFF_WMAT + (kb+1)*32 + ncol]      };
  v2f bHi = { C[OFF_WMAT + kb*32 + ncol + 16], C[OFF_WMAT + (kb+1)*32 + ncol + 16] };

  float* Hw = &Hs[wv*32*HSTRIDE];
  float accZ = 0.0f, accY = 0.0f;

  for (int t = gw; t < nTiles; t += nw){
    const int s = (t<<5) + lane;                 // this lane's own sample
    const float y1v = p.y1[s], y2v = p.y2[s];
    const float2 n1v = *(const float2*)(p.n1 + 2*s);
    const float2 n2v = *(const float2*)(p.n2 + 2*s);
    const float2 n3v = *(const float2*)(p.n3 + 2*s);
    const float2 n4v = *(const float2*)(p.n4 + 2*s);
    __builtin_prefetch(p.y1 + s + nw*32, 0, 0);  // speculative global_prefetch_b8

    // A tiles (16x4 f32): lo-lanes hold {K0,K1}={y1,y2}, hi-lanes {K2,K3}={1,0}
    const float y1o = __shfl_down(y1v, 16);
    const float y2o = __shfl_down(y2v, 16);
    v2f a0 = { hiH ? 1.0f : y1v, hiH ? 0.0f : y2v };   // samples t*32+0..15
    v2f a1 = { hiH ? 1.0f : y1o, hiH ? 0.0f : y2o };   // samples t*32+16..31
    v8f cz = {0.f,0.f,0.f,0.f,0.f,0.f,0.f,0.f};

    v8f d00 = __builtin_amdgcn_wmma_f32_16x16x4_f32(false,a0,false,bLo,(short)0,cz,false,false);
    v8f d01 = __builtin_amdgcn_wmma_f32_16x16x4_f32(false,a0,false,bHi,(short)0,cz,false,false);
    v8f d10 = __builtin_amdgcn_wmma_f32_16x16x4_f32(false,a1,false,bLo,(short)0,cz,false,false);
    v8f d11 = __builtin_amdgcn_wmma_f32_16x16x4_f32(false,a1,false,bHi,(short)0,cz,false,false);

    // D -> LDS transpose: row = in-wave sample, col = hidden unit (stride 36)
    const int mb = hiH ? 8 : 0;
    #pragma unroll
    for (int v=0; v<8; ++v){
      Hw[(mb+v)*HSTRIDE      + ncol     ] = d00[v];
      Hw[(mb+v)*HSTRIDE      + ncol + 16] = d01[v];
      Hw[(16+mb+v)*HSTRIDE   + ncol     ] = d10[v];
      Hw[(16+mb+v)*HSTRIDE   + ncol + 16] = d11[v];
    }
    __builtin_amdgcn_wave_barrier();
    asm volatile("s_wait_dscnt 0" ::: "memory");   // same-wave DS in-order; make explicit
    __builtin_amdgcn_wave_barrier();

    const float4* rp = (const float4*)(Hw + lane*HSTRIDE);
    const float4 q0=rp[0], q1=rp[1], q2=rp[2], q3=rp[3];
    const float4 q4=rp[4], q5=rp[5], q6=rp[6], q7=rp[7];
    __builtin_amdgcn_wave_barrier();

    // ---- inf0 -> z1 ----
    float li0, li1; dot8(q0,q1, w2i0, b2i00,b2i01, li0,li1);
    const bool bz1a = sigf(li0) > n1v.x, bz1b = sigf(li1) > n1v.y;
    const float z1a = bz1a?1.f:0.f,  z1b = bz1b?1.f:0.f;
    // ---- inf2 -> z3 ----
    float lc0, lc1; dot8(q2,q3, w2i2, b2i20,b2i21, lc0,lc1);
    const bool bz3a = sigf(lc0) > n3v.x, bz3b = sigf(lc1) > n3v.y;
    const float z3a = bz3a?1.f:0.f,  z3b = bz3b?1.f:0.f;
    // ---- g3 -> l2 ----
    float l20, l21; dot8(q4,q5, w2g3, b2g30,b2g31, l20,l21);
    // ---- inf1 (y2-part from WMMA, add z1 rows) -> z2 ----
    float lq0, lq1; dot8z(q6,q7, w2i1, b2i10,b2i11, wza,wzb, z1a,z1b, lq0,lq1);
    const bool bz2a = sigf(lq0) > n2v.x, bz2b = sigf(lq1) > n2v.y;
    const float z2a = bz2a?1.f:0.f,  z2b = bz2b?1.f:0.f;
    // ---- z4 from table ----
    const float p4a = sel4f(bz3a,bz3b, tp40[0],tp40[1],tp40[2],tp40[3]);
    const float p4b = sel4f(bz3a,bz3b, tp41[0],tp41[1],tp41[2],tp41[3]);
    const bool bz4a = p4a > n4v.x, bz4b = p4b > n4v.y;
    const float z4a = bz4a?1.f:0.f,  z4b = bz4b?1.f:0.f;

    // ---- losses (variable parts; constants folded into finalize) ----
    float bce1 = sel4f(bz2a,bz2b, sl1v[0],sl1v[1],sl1v[2],sl1v[3])
               - z1a*sel4f(bz2a,bz2b, tl10[0],tl10[1],tl10[2],tl10[3])
               - z1b*sel4f(bz2a,bz2b, tl11[0],tl11[1],tl11[2],tl11[3]);
    float bce2 = spabs(l20) - l20*z2a + spabs(l21) - l21*z2b;
    float bce3 = sel4f(bz4a,bz4b, sl3v[0],sl3v[1],sl3v[2],sl3v[3])
               - z3a*sel4f(bz4a,bz4b, tl30[0],tl30[1],tl30[2],tl30[3])
               - z3b*sel4f(bz4a,bz4b, tl31[0],tl31[1],tl31[2],tl31[3]);
    float bce4v = -(l4_0*z4a + l4_1*z4b);
    accZ += bce1 + bce2 + bce3 + bce4v;

    const float mu1 = sel4f(bz1a,bz1b, tm1[0],tm1[1],tm1[2],tm1[3]);
    const float mu2 = sel4f(bz3a,bz3b, tm2[0],tm2[1],tm2[2],tm2[3]);
    const float e1 = y1v-mu1, e2 = y2v-mu2;
    accY += 2.0f*(e1*e1 + e2*e2);
  }
  #undef LD4

  // wave reduce -> block reduce -> global atomics
  #pragma unroll
  for (int off=16; off; off>>=1){
    accZ += __shfl_xor(accZ, off);
    accY += __shfl_xor(accY, off);
  }
  if (lane==0){ atomicAdd(&redZ, accZ); atomicAdd(&redY, accY); }
  __syncthreads();
  if (threadIdx.x==0){ atomicAdd(&p.acc[0], redZ); atomicAdd(&p.acc[1], redY); }
}

__global__ void hm_fin(const float* ws, float invB, float* out){
  const float* C = ws + 16;
  // per-sample constants: bce4 softplus part; normal-logprob constant (2 terms)
  float c4 = spabs(C[OFF_L4+0]) + spabs(C[OFF_L4+1]);
  float lz = ws[0]*invB + c4;
  float ly = ws[1]*invB + 0.45158270528945485f;  // 2*(0.5*log(2pi) + log(0.5))
  out[0] = lz + ly; out[1] = lz; out[2] = ly;
}

extern "C" void kernel_launch(void* const* d_in, const int* in_sizes, int n_in,
                              void* d_out, int out_size, void* d_ws, size_t ws_size,
                              hipStream_t stream) {
  HmParams p;
  const float** f = (const float**)d_in;
  p.y1=f[0]; p.y2=f[1]; p.n1=f[2]; p.n2=f[3]; p.n3=f[4]; p.n4=f[5];
  p.inf0_w1=f[6];  p.inf0_b1=f[7];  p.inf0_w2=f[8];  p.inf0_b2=f[9];
  p.inf1_w1=f[10]; p.inf1_b1=f[11]; p.inf1_w2=f[12]; p.inf1_b2=f[13];
  p.inf2_w1=f[14]; p.inf2_b1=f[15]; p.inf2_w2=f[16]; p.inf2_b2=f[17];
  p.inf3_w1=f[18]; p.inf3_b1=f[19]; p.inf3_w2=f[20]; p.inf3_b2=f[21];
  p.g1_w1=f[22]; p.g1_b1=f[23]; p.g1_w2=f[24]; p.g1_b2=f[25];
  p.g2_w1=f[26]; p.g2_b1=f[27]; p.g2_w2=f[28]; p.g2_b2=f[29];
  p.g3_w1=f[30]; p.g3_b1=f[31]; p.g3_w2=f[32]; p.g3_b2=f[33];
  p.g4_w1=f[34]; p.g4_b1=f[35]; p.g4_w2=f[36]; p.g4_b2=f[37];
  p.g5_w1=f[38]; p.g5_b1=f[39]; p.g5_w2=f[40]; p.g5_b2=f[41];
  p.g0_w=f[42];
  p.ws = (float*)d_ws;
  p.B  = in_sizes[0];

  MainArgs m;
  m.y1=p.y1; m.y2=p.y2; m.n1=p.n1; m.n2=p.n2; m.n3=p.n3; m.n4=p.n4;
  m.cst = p.ws + 16;
  m.acc = p.ws;
  m.B   = p.B;

  const int nTiles = p.B >> 5;
  int blocks = (nTiles + WPB - 1) / WPB;
  if (blocks > 2048) blocks = 2048;
  if (blocks < 1) blocks = 1;

  hm_prep<<<1, 1, 0, stream>>>(p);
  hm_main<<<blocks, 256, 0, stream>>>(m);
  hm_fin<<<1, 1, 0, stream>>>(p.ws, 1.0f/(float)p.B, (float*)d_out);
}